// WarpMeshRasterizer_60687887892924
// MI455X (gfx1250) — compile-verified
//
#include <hip/hip_runtime.h>
#include <math.h>

// ---------------------------------------------------------------------------
// WarpMeshRasterizer for MI455X (gfx1250)
//  - fp32 WMMA 16x16x4 evaluates edge functions: D[m][n] = px*a + py*b + 1*c
//  - per-wave: 16 pixels (M rows), 16 triangles (N cols) per WMMA pair
//  - triangle records double-buffered in LDS via GLOBAL_LOAD_ASYNC_TO_LDS_B128
//  - z-buffer semantics: min z, tie -> smallest triangle id (== reference scan)
// ---------------------------------------------------------------------------

typedef float v2f __attribute__((ext_vector_type(2)));
typedef float v8f __attribute__((ext_vector_type(8)));

#define BIGF 10000000000.0f

// Triangle record: 16 floats, 64B
// [0]=a0 [1]=b0 [2]=c0 [3]=0   (w0 = a0*px + b0*py + c0)
// [4]=a1 [5]=b1 [6]=c1 [7]=0   (w1 = a1*px + b1*py + c1)
// [8]=z0 [9]=z1 [10]=z2 [11]=0
// [12..15] = 0
// Zero slots at +3/+7 feed the K=3 row of the WMMA B operand.

__global__ void wmr_setup_kernel(const float* __restrict__ verts,
                                 const int*   __restrict__ tri,
                                 const float* __restrict__ mvp,
                                 int W, int T, int Tpad,
                                 float* __restrict__ rec) {
    int t = blockIdx.x * blockDim.x + threadIdx.x;
    if (t >= Tpad) return;

    // default "fail" record: w0 = -1 everywhere -> never inside
    float a0 = 0.f, b0 = 0.f, c0 = -1.f;
    float a1 = 0.f, b1 = 0.f, c1 = 0.f;
    float z0 = 0.f, z1 = 0.f, z2 = 0.f;

    if (t < T) {
        int xs[3], ys[3];
        float zs[3];
#pragma unroll
        for (int k = 0; k < 3; ++k) {
            int vi = tri[t * 3 + k];
            float vx = verts[vi * 3 + 0];
            float vy = verts[vi * 3 + 1];
            float vz = verts[vi * 3 + 2];
            // clip = posw @ mvp.T  (mvp row-major 4x4)
            float p0 = mvp[0]  * vx + mvp[1]  * vy + mvp[2]  * vz + mvp[3];
            float p1 = mvp[4]  * vx + mvp[5]  * vy + mvp[6]  * vz + mvp[7];
            float p2 = mvp[8]  * vx + mvp[9]  * vy + mvp[10] * vz + mvp[11];
            float p3 = mvp[12] * vx + mvp[13] * vy + mvp[14] * vz + mvp[15];
            if (p3 != 0.0f) { p0 /= p3; p1 /= p3; p2 /= p3; }
            float xf = truncf((p0 + 1.0f) * 0.5f * (float)W);
            float yf = truncf((p1 + 1.0f) * 0.5f * (float)W);
            xs[k] = (int)xf;
            ys[k] = (int)yf;
            zs[k] = p2;
        }
        int A0 = ys[1] - ys[2], B0 = xs[2] - xs[1];
        int A1 = ys[2] - ys[0], B1 = xs[0] - xs[2];
        int denom = A0 * (xs[0] - xs[2]) + B0 * (ys[0] - ys[2]);
        if (denom != 0) {  // |float(denom)| >= 1e-8  <=>  denom != 0 (integer)
            float safe = (float)denom;
            int C0 = -(A0 * xs[2] + B0 * ys[2]);
            int C1 = -(A1 * xs[2] + B1 * ys[2]);
            a0 = (float)A0 / safe; b0 = (float)B0 / safe; c0 = (float)C0 / safe;
            a1 = (float)A1 / safe; b1 = (float)B1 / safe; c1 = (float)C1 / safe;
            z0 = zs[0]; z1 = zs[1]; z2 = zs[2];
        }
    }
    float4* dst = (float4*)(rec + (size_t)t * 16);
    dst[0] = make_float4(a0, b0, c0, 0.f);
    dst[1] = make_float4(a1, b1, c1, 0.f);
    dst[2] = make_float4(z0, z1, z2, 0.f);
    dst[3] = make_float4(0.f, 0.f, 0.f, 0.f);
}

// Issue async copy of one 16KB stage (256 records) into LDS buffer.
// Each of the 256 threads moves 4 x 16B, VGPR-bypassing, tracked by ASYNCcnt.
__device__ __forceinline__ void stage_async(const float* __restrict__ rec,
                                            int s, unsigned ldsBase, int tix) {
#pragma unroll
    for (int k = 0; k < 4; ++k) {
        const void* g = (const void*)(rec + (size_t)s * 4096 + (size_t)k * 1024 + tix * 4);
        unsigned l = ldsBase + (unsigned)(tix * 16 + k * 4096);
        asm volatile("global_load_async_to_lds_b128 %0, %1, off"
                     :: "v"(l), "v"(g) : "memory");
    }
}

__global__ __launch_bounds__(256)
void wmr_raster_kernel(const float* __restrict__ rec,
                       float* __restrict__ out,
                       int W, int Tpad) {
    __shared__ __align__(16) float buf[2][256 * 16];   // double-buffered records
    __shared__ __align__(16) float obuf[8][16][4];     // per-wave publish buffer

    const int tix  = threadIdx.x;
    const int lane = tix & 31;
    const int wave = tix >> 5;
    const int x0 = blockIdx.x * 16;
    const int py = blockIdx.y * 8 + wave;
    const bool lowHalf = (lane < 16);
    const int col = lane & 15;

    const unsigned ldsBase0 = (unsigned)(size_t)(&buf[0][0]);
    const unsigned ldsBase1 = (unsigned)(size_t)(&buf[1][0]);

    // A matrix (16x4 f32): row m = (px, py, 1, 0)
    // lanes 0-15: vgpr0=K0(px), vgpr1=K1(py); lanes 16-31: vgpr0=K2(1), vgpr1=K3(0)
    v2f a;
    a.x = lowHalf ? (float)(x0 + lane) : 1.0f;
    a.y = lowHalf ? (float)py : 0.0f;

    // Contiguous B fetch: low half reads (a,b) at +0; high half reads (c,0) at +2
    const int boff = lowHalf ? 0 : 2;

    float bz[8], bw0[8], bw1[8];
    int bt[8];
#pragma unroll
    for (int i = 0; i < 8; ++i) { bz[i] = BIGF; bw0[i] = 0.0f; bw1[i] = 0.0f; bt[i] = -1; }

    const int nStages = Tpad >> 8;   // 256 triangles per stage
    stage_async(rec, 0, ldsBase0, tix);

    for (int s = 0; s < nStages; ++s) {
        // my copies for stage s are done ...
        asm volatile("s_wait_asynccnt 0x0" ::: "memory");
        // ... everyone's copies are done AND everyone finished reading the
        // buffer we are about to overwrite (stage s-1's buffer)
        __syncthreads();
        if (s + 1 < nStages)
            stage_async(rec, s + 1, ((s + 1) & 1) ? ldsBase1 : ldsBase0, tix);

        const float* bufS = &buf[s & 1][0];
        for (int c = 0; c < 16; ++c) {   // 16 triangles per WMMA pair
            const float* tb = bufS + c * 256 + col * 16;
            v2f b0v = *(const v2f*)(tb + 0 + boff);  // (a0,b0) | (c0,0)
            v2f b1v = *(const v2f*)(tb + 4 + boff);  // (a1,b1) | (c1,0)
            float4 zv = *(const float4*)(tb + 8);    // (z0,z1,z2,0)
            int tidc = s * 256 + c * 16 + col;

            v8f cz = {};
            // D1[m][n] = w0(pixel m, triangle n); D2 -> w1
            v8f d1 = __builtin_amdgcn_wmma_f32_16x16x4_f32(
                false, a, false, b0v, (short)0, cz, false, false);
            v8f d2 = __builtin_amdgcn_wmma_f32_16x16x4_f32(
                false, a, false, b1v, (short)0, cz, false, false);

#pragma unroll
            for (int rr = 0; rr < 8; ++rr) {
                float w0 = d1[rr], w1 = d2[rr];
                float w2 = 1.0f - w0 - w1;
                float mn = fminf(fminf(w0, w1), w2);
                float zpix = w0 * zv.x + w1 * zv.y + w2 * zv.z;
                float zm = (mn >= 0.0f) ? zpix : BIGF;
                bool take = zm < bz[rr];   // tid strictly increases per lane
                bz[rr]  = take ? zm   : bz[rr];
                bt[rr]  = take ? tidc : bt[rr];
                bw0[rr] = take ? w0   : bw0[rr];
                bw1[rr] = take ? w1   : bw1[rr];
            }
        }
    }

    // reduce across the 16 triangle columns inside each half-wave
    // rule: smaller z wins; equal z -> smaller tid (== reference scan semantics)
#pragma unroll
    for (int rr = 0; rr < 8; ++rr) {
#pragma unroll
        for (int sft = 1; sft < 16; sft <<= 1) {
            float oz = __shfl_xor(bz[rr], sft, 32);
            int   ot = __shfl_xor(bt[rr], sft, 32);
            float o0 = __shfl_xor(bw0[rr], sft, 32);
            float o1 = __shfl_xor(bw1[rr], sft, 32);
            bool take = (oz < bz[rr]) || ((oz == bz[rr]) && (ot < bt[rr]));
            bz[rr]  = take ? oz : bz[rr];
            bt[rr]  = take ? ot : bt[rr];
            bw0[rr] = take ? o0 : bw0[rr];
            bw1[rr] = take ? o1 : bw1[rr];
        }
    }

    // lane 0 publishes rows 0..7, lane 16 publishes rows 8..15
    if ((lane & 15) == 0) {
        int rowBase = (lane >> 4) * 8;
#pragma unroll
        for (int rr = 0; rr < 8; ++rr) {
            bool hit = bt[rr] >= 0;
            float b2 = hit ? (1.0f - bw0[rr] - bw1[rr]) : 0.0f;
            obuf[wave][rowBase + rr][0] = bw0[rr];
            obuf[wave][rowBase + rr][1] = bw1[rr];
            obuf[wave][rowBase + rr][2] = b2;
            obuf[wave][rowBase + rr][3] = (float)(bt[rr] + 1);
        }
    }
    __syncthreads();

    if (lane < 16) {
        float4 v = *(const float4*)&obuf[wave][lane][0];
        int px = x0 + lane;
        *(float4*)(out + ((size_t)py * W + px) * 4) = v;
    }
}

extern "C" void kernel_launch(void* const* d_in, const int* in_sizes, int n_in,
                              void* d_out, int out_size, void* d_ws, size_t ws_size,
                              hipStream_t stream) {
    const float* verts = (const float*)d_in[0];
    const int*   tri   = (const int*)d_in[1];
    const float* mvp   = (const float*)d_in[2];
    // resolution: out has W*W*4 floats
    int W = (int)lround(sqrt((double)out_size / 4.0));
    int T = in_sizes[1] / 3;
    int Tpad = ((T + 255) / 256) * 256;

    float* rec = (float*)d_ws;  // Tpad * 16 floats (64B per triangle)

    wmr_setup_kernel<<<(Tpad + 127) / 128, 128, 0, stream>>>(
        verts, tri, mvp, W, T, Tpad, rec);

    dim3 grid(W / 16, W / 8);
    wmr_raster_kernel<<<grid, 256, 0, stream>>>(rec, (float*)d_out, W, Tpad);
}